// GAT_26714696581831
// MI455X (gfx1250) — compile-verified
//
#include <hip/hip_runtime.h>
#include <hip/hip_bf16.h>

typedef __attribute__((ext_vector_type(2))) float v2f;
typedef __attribute__((ext_vector_type(8))) float v8f;

#define NNODES 50000
#define NEDGES 800000
#define FIN    128
#define HIDC   64
#define NHEADS 4
#define NOUTC  2
#define NEG_SLOPE 0.2f

// ---------------------------------------------------------------------------
// fp32 WMMA GEMM: C[M x P] = A[M x K] @ W[K x P]
// One wave computes a 16x64 tile using 4x V_WMMA_F32_16X16X4_F32 per k-step.
// Requires: M % 16 == 0, K % 4 == 0, P % 64 == 0. EXEC all-ones per wave.
// A fragment layout (16x4): lanes 0-15 hold K=k,k+1; lanes 16-31 hold K=k+2,k+3; M = lane%16.
// B fragment layout (4x16): mirrored; N = lane%16.
// C layout: vgpr r -> row = 16*tile + r + 8*(lane/16), col = lane%16.
// ---------------------------------------------------------------------------
__global__ __launch_bounds__(256) void gemm_wmma_f32(
    const float* __restrict__ A, const float* __restrict__ W,
    float* __restrict__ C, int M, int K, int P)
{
    int wave = (blockIdx.x * blockDim.x + threadIdx.x) >> 5;
    int lane = threadIdx.x & 31;
    int nCG  = P >> 6;               // 64-wide column groups
    int nRT  = M >> 4;
    if (wave >= nRT * nCG) return;   // wave-uniform: EXEC stays all-ones
    int rowTile = wave / nCG;
    int cg      = wave - rowTile * nCG;

    int half  = lane >> 4;           // 0 or 1
    int l15   = lane & 15;
    int kk    = half * 2;            // K sub-offset for A/B fragments
    int arow  = rowTile * 16 + l15;
    int cbase = cg * 64 + l15;

    v8f acc0 = {}, acc1 = {}, acc2 = {}, acc3 = {};
    const float* Arow = A + (long)arow * K;
    for (int k = 0; k < K; k += 4) {
        v2f a;
        a.x = Arow[k + kk];
        a.y = Arow[k + kk + 1];
        const float* W0 = W + (long)(k + kk)     * P + cbase;
        const float* W1 = W + (long)(k + kk + 1) * P + cbase;
        v2f b0, b1, b2, b3;
        b0.x = W0[0];  b0.y = W1[0];
        b1.x = W0[16]; b1.y = W1[16];
        b2.x = W0[32]; b2.y = W1[32];
        b3.x = W0[48]; b3.y = W1[48];
        acc0 = __builtin_amdgcn_wmma_f32_16x16x4_f32(false, a, false, b0, (short)0, acc0, false, false);
        acc1 = __builtin_amdgcn_wmma_f32_16x16x4_f32(false, a, false, b1, (short)0, acc1, false, false);
        acc2 = __builtin_amdgcn_wmma_f32_16x16x4_f32(false, a, false, b2, (short)0, acc2, false, false);
        acc3 = __builtin_amdgcn_wmma_f32_16x16x4_f32(false, a, false, b3, (short)0, acc3, false, false);
    }
    int orow0 = rowTile * 16 + half * 8;
    #pragma unroll
    for (int r = 0; r < 8; ++r) {
        long rb = (long)(orow0 + r) * P + cg * 64 + l15;
        C[rb +  0] = acc0[r];
        C[rb + 16] = acc1[r];
        C[rb + 32] = acc2[r];
        C[rb + 48] = acc3[r];
    }
}

// a_src[n,h] = sum_c xs[n,h*64+c]*as[h*64+c] ; same for a_dst from xd/ad (layer 1)
__global__ void att_reduce1(const float* __restrict__ xs, const float* __restrict__ xd,
                            const float* __restrict__ as, const float* __restrict__ ad,
                            float* __restrict__ a_src, float* __restrict__ a_dst)
{
    int idx = blockIdx.x * blockDim.x + threadIdx.x;   // n*4 + h
    if (idx >= NNODES * NHEADS) return;
    int n = idx >> 2, h = idx & 3;
    const float* xr  = xs + (long)n * (NHEADS * HIDC) + h * HIDC;
    const float* xrd = xd + (long)n * (NHEADS * HIDC) + h * HIDC;
    const float* ar  = as + h * HIDC;
    const float* ard = ad + h * HIDC;
    float ss = 0.f, sd = 0.f;
    #pragma unroll 4
    for (int c = 0; c < HIDC; ++c) { ss += xr[c] * ar[c]; sd += xrd[c] * ard[c]; }
    a_src[idx] = ss;
    a_dst[idx] = sd;
}

__global__ void fill_kernel(float* __restrict__ p, float v, int n)
{
    int i = blockIdx.x * blockDim.x + threadIdx.x;
    if (i < n) p[i] = v;
}

__device__ __forceinline__ float leaky(float v) { return v > 0.f ? v : NEG_SLOPE * v; }

// float max via native integer atomics (no CAS loop)
__device__ __forceinline__ void atomicMaxFloat(float* addr, float v)
{
    if (v >= 0.f) atomicMax((int*)addr, __float_as_int(v));
    else          atomicMin((unsigned int*)addr, __float_as_uint(v));
}

__global__ void edge_max(const int* __restrict__ src, const int* __restrict__ dst,
                         const float* __restrict__ a_src, const float* __restrict__ a_dst,
                         float* __restrict__ m)
{
    int e = blockIdx.x * blockDim.x + threadIdx.x;
    if (e >= NEDGES) return;
    int s = src[e], d = dst[e];
    #pragma unroll
    for (int h = 0; h < NHEADS; ++h) {
        float v = leaky(a_src[s * 4 + h] + a_dst[d * 4 + h]);
        atomicMaxFloat(&m[d * 4 + h], v);
    }
}

__global__ void edge_expsum(const int* __restrict__ src, const int* __restrict__ dst,
                            const float* __restrict__ a_src, const float* __restrict__ a_dst,
                            const float* __restrict__ m,
                            float* __restrict__ ex, float* __restrict__ ssum)
{
    int e = blockIdx.x * blockDim.x + threadIdx.x;
    if (e >= NEDGES) return;
    int s = src[e], d = dst[e];
    #pragma unroll
    for (int h = 0; h < NHEADS; ++h) {
        float v = leaky(a_src[s * 4 + h] + a_dst[d * 4 + h]);
        float x = __expf(v - m[d * 4 + h]);
        ex[e * 4 + h] = x;
        atomicAdd(&ssum[d * 4 + h], x);
    }
}

// s[i] = scale / (s[i] + 1e-16) ; scale folds the 1/HEADS mean into alpha
__global__ void rcp_scale(float* __restrict__ s, float scale, int n)
{
    int i = blockIdx.x * blockDim.x + threadIdx.x;
    if (i < n) s[i] = scale / (s[i] + 1e-16f);
}

// agg[dst,c] += sum_h alpha[e,h] * feat[src, h*CH + c]   (one atomic per (e,c))
template <int CH>
__global__ void edge_scatter(const int* __restrict__ src, const int* __restrict__ dst,
                             const float* __restrict__ ex, const float* __restrict__ sr,
                             const float* __restrict__ feat, float* __restrict__ agg)
{
    long idx = (long)blockIdx.x * blockDim.x + threadIdx.x;
    if (idx >= (long)NEDGES * CH) return;
    int e = (int)(idx / CH);
    int c = (int)(idx - (long)e * CH);
    int s = src[e], d = dst[e];
    float val = 0.f;
    #pragma unroll
    for (int h = 0; h < NHEADS; ++h) {
        float alpha = ex[e * 4 + h] * sr[d * 4 + h];
        val += alpha * feat[(long)s * (NHEADS * CH) + h * CH + c];
    }
    atomicAdd(&agg[(long)d * CH + c], val);
}

// h = sigmoid(agg + b1 + hskip + bl1)
__global__ void combine1(const float* __restrict__ agg, const float* __restrict__ hskip,
                         const float* __restrict__ b1, const float* __restrict__ bl1,
                         float* __restrict__ hout)
{
    int idx = blockIdx.x * blockDim.x + threadIdx.x;
    if (idx >= NNODES * HIDC) return;
    int c = idx & (HIDC - 1);
    float v = agg[idx] + hskip[idx] + b1[c] + bl1[c];
    hout[idx] = 1.f / (1.f + __expf(-v));
}

// Layer-2 projections: xs2 = h@W2s, a_src2/a_dst2 reductions, hskip2 = h@Wl2 + bl2
__global__ void layer2_linear(const float* __restrict__ h,
                              const float* __restrict__ W2s, const float* __restrict__ W2d,
                              const float* __restrict__ a2s, const float* __restrict__ a2d,
                              const float* __restrict__ Wl2, const float* __restrict__ bl2,
                              float* __restrict__ xs2, float* __restrict__ a_src2,
                              float* __restrict__ a_dst2, float* __restrict__ hskip2)
{
    int n = blockIdx.x * blockDim.x + threadIdx.x;
    if (n >= NNODES) return;
    const float* hr = h + (long)n * HIDC;
    float xs[8] = {0,0,0,0,0,0,0,0};
    float xd[8] = {0,0,0,0,0,0,0,0};
    float sk0 = 0.f, sk1 = 0.f;
    for (int k = 0; k < HIDC; ++k) {
        float hv = hr[k];
        #pragma unroll
        for (int j = 0; j < 8; ++j) {
            xs[j] += hv * W2s[k * 8 + j];
            xd[j] += hv * W2d[k * 8 + j];
        }
        sk0 += hv * Wl2[k * 2 + 0];
        sk1 += hv * Wl2[k * 2 + 1];
    }
    #pragma unroll
    for (int j = 0; j < 8; ++j) xs2[n * 8 + j] = xs[j];
    #pragma unroll
    for (int hh = 0; hh < NHEADS; ++hh) {
        a_src2[n * 4 + hh] = xs[hh * 2] * a2s[hh * 2] + xs[hh * 2 + 1] * a2s[hh * 2 + 1];
        a_dst2[n * 4 + hh] = xd[hh * 2] * a2d[hh * 2] + xd[hh * 2 + 1] * a2d[hh * 2 + 1];
    }
    hskip2[n * 2 + 0] = sk0 + bl2[0];
    hskip2[n * 2 + 1] = sk1 + bl2[1];
}

__global__ void final_out(const float* __restrict__ agg2, const float* __restrict__ hskip2,
                          const float* __restrict__ b2, float* __restrict__ out)
{
    int idx = blockIdx.x * blockDim.x + threadIdx.x;
    if (idx >= NNODES * NOUTC) return;
    int c = idx & 1;
    out[idx] = agg2[idx] + hskip2[idx] + b2[c];
}

extern "C" void kernel_launch(void* const* d_in, const int* in_sizes, int n_in,
                              void* d_out, int out_size, void* d_ws, size_t ws_size,
                              hipStream_t stream)
{
    const float* x   = (const float*)d_in[0];
    const int*   ei  = (const int*)d_in[1];
    const float* W1s = (const float*)d_in[2];
    const float* W1d = (const float*)d_in[3];
    const float* a1s = (const float*)d_in[4];
    const float* a1d = (const float*)d_in[5];
    const float* b1  = (const float*)d_in[6];
    const float* Wl1 = (const float*)d_in[7];
    const float* bl1 = (const float*)d_in[8];
    const float* W2s = (const float*)d_in[9];
    const float* W2d = (const float*)d_in[10];
    const float* a2s = (const float*)d_in[11];
    const float* a2d = (const float*)d_in[12];
    const float* b2  = (const float*)d_in[13];
    const float* Wl2 = (const float*)d_in[14];
    const float* bl2 = (const float*)d_in[15];
    const int* srcv = ei;
    const int* dstv = ei + NEDGES;

    const long N = NNODES, E = NEDGES;
    float* base   = (float*)d_ws;
    float* xs1    = base;                 // N*256
    float* hskip1 = xs1 + N * 256;        // N*64
    float* asrc1  = hskip1 + N * 64;      // N*4
    float* adst1  = asrc1 + N * 4;        // N*4
    float* mbuf   = adst1 + N * 4;        // N*4   (both layers)
    float* sbuf   = mbuf + N * 4;         // N*4   (both layers)
    float* exbuf  = sbuf + N * 4;         // E*4   (both layers)
    float* agg1   = exbuf + E * 4;        // N*64
    float* xd1    = agg1 + N * 64;        // N*256 (dead after att_reduce1; aliased below)
    float* hbuf   = xd1;                  // N*64  (layer-1 output)
    float* xs2    = xd1 + N * 64;         // N*8
    float* asrc2  = xs2 + N * 8;          // N*4
    float* adst2  = asrc2 + N * 4;        // N*4
    float* hskip2 = adst2 + N * 4;        // N*2
    float* agg2   = hskip2 + N * 2;       // N*2

    const int B = 256;
    // ---- layer 1 GEMMs (fp32 WMMA) ----
    int waves256 = (NNODES / 16) * (256 / 64);   // 12500 waves
    int waves64  = (NNODES / 16) * (64 / 64);    // 3125 waves
    gemm_wmma_f32<<<(waves256 * 32 + B - 1) / B, B, 0, stream>>>(x, W1s, xs1, NNODES, FIN, 256);
    gemm_wmma_f32<<<(waves256 * 32 + B - 1) / B, B, 0, stream>>>(x, W1d, xd1, NNODES, FIN, 256);
    gemm_wmma_f32<<<(waves64  * 32 + B - 1) / B, B, 0, stream>>>(x, Wl1, hskip1, NNODES, FIN, 64);
    att_reduce1<<<(NNODES * 4 + B - 1) / B, B, 0, stream>>>(xs1, xd1, a1s, a1d, asrc1, adst1);

    // ---- layer 1 edge softmax + aggregate ----
    fill_kernel<<<(NNODES * 4 + B - 1) / B, B, 0, stream>>>(mbuf, -1e30f, NNODES * 4);
    fill_kernel<<<(NNODES * 4 + B - 1) / B, B, 0, stream>>>(sbuf, 0.f, NNODES * 4);
    fill_kernel<<<(NNODES * 64 + B - 1) / B, B, 0, stream>>>(agg1, 0.f, NNODES * 64);
    edge_max<<<(NEDGES + B - 1) / B, B, 0, stream>>>(srcv, dstv, asrc1, adst1, mbuf);
    edge_expsum<<<(NEDGES + B - 1) / B, B, 0, stream>>>(srcv, dstv, asrc1, adst1, mbuf, exbuf, sbuf);
    rcp_scale<<<(NNODES * 4 + B - 1) / B, B, 0, stream>>>(sbuf, 0.25f, NNODES * 4);
    {
        long tot = E * 64;
        edge_scatter<64><<<(int)((tot + B - 1) / B), B, 0, stream>>>(srcv, dstv, exbuf, sbuf, xs1, agg1);
    }
    combine1<<<(NNODES * 64 + B - 1) / B, B, 0, stream>>>(agg1, hskip1, b1, bl1, hbuf);

    // ---- layer 2 projections ----
    layer2_linear<<<(NNODES + B - 1) / B, B, 0, stream>>>(hbuf, W2s, W2d, a2s, a2d, Wl2, bl2,
                                                          xs2, asrc2, adst2, hskip2);

    // ---- layer 2 edge softmax + aggregate ----
    fill_kernel<<<(NNODES * 4 + B - 1) / B, B, 0, stream>>>(mbuf, -1e30f, NNODES * 4);
    fill_kernel<<<(NNODES * 4 + B - 1) / B, B, 0, stream>>>(sbuf, 0.f, NNODES * 4);
    fill_kernel<<<(NNODES * 2 + B - 1) / B, B, 0, stream>>>(agg2, 0.f, NNODES * 2);
    edge_max<<<(NEDGES + B - 1) / B, B, 0, stream>>>(srcv, dstv, asrc2, adst2, mbuf);
    edge_expsum<<<(NEDGES + B - 1) / B, B, 0, stream>>>(srcv, dstv, asrc2, adst2, mbuf, exbuf, sbuf);
    rcp_scale<<<(NNODES * 4 + B - 1) / B, B, 0, stream>>>(sbuf, 0.25f, NNODES * 4);
    {
        long tot = E * 2;
        edge_scatter<2><<<(int)((tot + B - 1) / B), B, 0, stream>>>(srcv, dstv, exbuf, sbuf, xs2, agg2);
    }
    final_out<<<(NNODES * 2 + B - 1) / B, B, 0, stream>>>(agg2, hskip2, b2, (float*)d_out);
}